// GGUFLinear_9758165696668
// MI455X (gfx1250) — compile-verified
//
#include <hip/hip_runtime.h>

typedef __attribute__((ext_vector_type(2))) float v2f;
typedef __attribute__((ext_vector_type(4))) float v4f;
typedef __attribute__((ext_vector_type(8))) float v8f;
typedef __attribute__((ext_vector_type(2))) int   v2i;

#define TOKENS 16
#define IN     4096
#define OUT    11008
#define QBLK   32
#define NBLK   (IN / QBLK)      // 128 blocks per row
#define PAD    4
#define XROW   (IN + PAD)       // 4100 floats: bank-conflict-free row stride
#define WAVES  8                // 8 wave32 per workgroup
#define NTILES (OUT / 16)       // 688 N-tiles of 16 outputs

// One wave computes one 16(M) x 16(N) output tile over the full K=4096,
// using V_WMMA_F32_16X16X4_F32 with A from LDS-staged x and B dequantized
// on the fly from the Q8 codes (int32 storage) streaming from HBM.
__global__ __launch_bounds__(WAVES * 32)
void gguf_linear_wmma(const float* __restrict__ x,
                      const int*   __restrict__ qw,
                      const float* __restrict__ scales,
                      const float* __restrict__ bias,
                      float* __restrict__ out)
{
    extern __shared__ float xs[];   // TOKENS * XROW floats = 262,400 B (fits 320 KB LDS)

    const int tid = threadIdx.x;

    // ---- Stage ALL of x (16 x 4096 fp32 = 256 KB) into LDS, padded rows ----
    // Coalesced float4 global loads; 64 iterations per thread.
    for (int i = tid; i < (TOKENS * IN) / 4; i += WAVES * 32) {
        const int m = i >> 10;            // i / (IN/4)
        const int k = (i & 1023) << 2;    // (i % 1024) * 4
        v4f v = *(const v4f*)(x + m * IN + k);
        *(v4f*)(xs + m * XROW + k) = v;
    }
    __syncthreads();

    const int wave = tid >> 5;
    const int lane = tid & 31;
    const int tile = blockIdx.x * WAVES + wave;   // 86 blocks * 8 waves = 688 = NTILES
    if (tile >= NTILES) return;                   // wave-uniform; never taken at this grid

    // Fragment geometry for 32-bit 16x16x4 WMMA (ISA 7.12.2):
    //   lane<16 : m/n = lane,     K-offset 0 (holds K=k0,k0+1 in the 2 VGPRs)
    //   lane>=16: m/n = lane-16,  K-offset 2 (holds K=k0+2,k0+3)
    const int row    = lane & 15;
    const int koff   = (lane >> 4) << 1;          // 0 or 2
    const int n      = tile * 16 + row;

    const float* xr = xs + row * XROW + koff;               // A source (LDS)
    const int*   qr = qw + (size_t)n * IN + koff;           // B codes (HBM stream)
    const float* sr = scales + (size_t)n * NBLK;            // per-block scales

    v8f c = {0.f, 0.f, 0.f, 0.f, 0.f, 0.f, 0.f, 0.f};

    for (int kb = 0; kb < NBLK; ++kb) {
        const float s   = sr[kb];
        const float off = -128.0f * s;

        // Prefetch the qweight stream ~4 blocks (512 B/row) ahead into GL2.
        __builtin_prefetch(qr + (kb + 4) * QBLK, 0, 1);

        const int kbase = kb * QBLK;
#pragma unroll
        for (int i = 0; i < 8; ++i) {
            const int kk = kbase + i * 4;
            // Each lane: 8 B of codes (its 2 K slots); lanes L and L+16 cover
            // 16 contiguous bytes, the 8-step unroll covers the 128 B line.
            v2i q = *(const v2i*)(qr + kk);          // global_load_b64
            v2f b;
            b.x = (float)q.x * s + off;              // (q-128)*scale, exact fp32
            b.y = (float)q.y * s + off;
            v2f a = *(const v2f*)(xr + kk);          // ds_load_b64, conflict-free
            c = __builtin_amdgcn_wmma_f32_16x16x4_f32(
                    /*neg_a=*/false, a, /*neg_b=*/false, b,
                    /*c_mod=*/(short)0, c, /*reuse_a=*/false, /*reuse_b=*/false);
        }
    }

    // ---- Epilogue: C/D layout -> lane holds N = row, VGPR r -> M = r (+8 if lane>=16)
    const float bz  = bias[n];
    const int   mo  = (lane >> 4) << 3;  // 0 or 8
#pragma unroll
    for (int r = 0; r < 8; ++r) {
        out[(size_t)(mo + r) * OUT + n] = c[r] + bz;
    }
}

extern "C" void kernel_launch(void* const* d_in, const int* in_sizes, int n_in,
                              void* d_out, int out_size, void* d_ws, size_t ws_size,
                              hipStream_t stream) {
    const float* x      = (const float*)d_in[0];
    const int*   qw     = (const int*)  d_in[1];
    const float* scales = (const float*)d_in[2];
    const float* bias   = (const float*)d_in[3];
    float*       out    = (float*)d_out;

    const size_t lds_bytes = (size_t)TOKENS * XROW * sizeof(float);  // 262,400 B
    dim3 grid(NTILES / WAVES);   // 86
    dim3 block(WAVES * 32);      // 256 threads = 8 wave32
    gguf_linear_wmma<<<grid, block, lds_bytes, stream>>>(x, qw, scales, bias, out);
}